// SAGELayer_53712861004559
// MI455X (gfx1250) — compile-verified
//
#include <hip/hip_runtime.h>
#include <hip/hip_bf16.h>

typedef float v2f __attribute__((ext_vector_type(2)));
typedef float v8f __attribute__((ext_vector_type(8)));

#define D_FEAT   64
#define D_K      128     // concat width
#define TILE_M   16      // rows per wave tile
#define WAVES_PB 8       // waves per block (256 threads)
#define LDS_WSTR 130     // padded LDS row stride (floats) to avoid bank conflicts

// ---------------------------------------------------------------------------
// Zero-init workspace: msg_sum (n_nodes*64) + deg (n_nodes)
// ---------------------------------------------------------------------------
__global__ void sage_init_kernel(float* __restrict__ ws, int total) {
    int i = blockIdx.x * blockDim.x + threadIdx.x;
    int stride = gridDim.x * blockDim.x;
    for (; i < total; i += stride) ws[i] = 0.0f;
}

// ---------------------------------------------------------------------------
// Edge kernel: msg = [h[src] || efeat] @ W_msg^T + b_msg, scatter-add to
// msg_sum[dst], accumulate deg[dst]. One wave handles 16 edges using
// V_WMMA_F32_16X16X4_F32 (M=16, N=16 x4 tiles, K=128 in 32 steps of 4).
// ---------------------------------------------------------------------------
__global__ void __launch_bounds__(256)
sage_edge_kernel(const float* __restrict__ nfeats,   // n_nodes x 64
                 const float* __restrict__ efeats,   // n_edges x 64
                 const int*   __restrict__ src,
                 const int*   __restrict__ dst,
                 const float* __restrict__ Wm,       // 64 x 128 row-major
                 const float* __restrict__ bm,       // 64
                 float*       __restrict__ msg_sum,  // n_nodes x 64
                 float*       __restrict__ deg,      // n_nodes
                 int n_edges)
{
    __shared__ float wlds[D_FEAT * LDS_WSTR];

    // Stage W_msg into LDS (padded rows).
    for (int i = threadIdx.x; i < D_FEAT * D_K; i += blockDim.x)
        wlds[(i >> 7) * LDS_WSTR + (i & 127)] = Wm[i];
    __syncthreads();

    const int wave = threadIdx.x >> 5;
    const int lane = threadIdx.x & 31;
    const int tile = blockIdx.x * WAVES_PB + wave;
    const int e0   = tile * TILE_M;
    if (e0 >= n_edges) return;           // wave-uniform: EXEC stays all-1s

    const int row  = lane & 15;          // matrix row this lane feeds (A frag)
    const int hi   = lane >> 4;          // 0: K%4 in {0,1}; 1: K%4 in {2,3}
    const int koff = hi * 2;

    int e_row = e0 + row;
    bool row_valid = (e_row < n_edges);
    if (!row_valid) e_row = n_edges - 1; // clamp for safe loads

    const int s = src[e_row];

    // Degree accumulation: one lane per edge.
    if (hi == 0 && row_valid) atomicAdd(&deg[dst[e_row]], 1.0f);

    // dst rows needed for the scatter of C rows m = r + hi*8
    int drow[8];
#pragma unroll
    for (int r = 0; r < 8; ++r) {
        int m = r + hi * 8;
        int e = e0 + m;
        drow[r] = (e < n_edges) ? dst[e] : -1;
    }

    // A fragments: X[row][k] ; k 0..63 from h[src], 64..127 from efeat.
    const float* hrow = nfeats + (size_t)s * D_FEAT;
    const float* erow = efeats + (size_t)e_row * D_FEAT;
    v2f a[32];
#pragma unroll
    for (int i = 0; i < 16; ++i)
        a[i] = *(const v2f*)(hrow + i * 4 + koff);
#pragma unroll
    for (int i = 0; i < 16; ++i)
        a[16 + i] = *(const v2f*)(erow + i * 4 + koff);
    if (!row_valid) {
#pragma unroll
        for (int i = 0; i < 32; ++i) a[i] = (v2f){0.0f, 0.0f};
    }

    // 4 N-tiles of 16 columns each.
#pragma unroll
    for (int n0 = 0; n0 < 4; ++n0) {
        const int ncol = n0 * 16 + row;       // C-layout: lane column
        const float bias = bm[ncol];
        v8f acc = {bias, bias, bias, bias, bias, bias, bias, bias};

        const float* wrow = wlds + ncol * LDS_WSTR + koff;
#pragma unroll
        for (int i = 0; i < 32; ++i) {
            v2f bfrag = *(const v2f*)(wrow + i * 4);
            acc = __builtin_amdgcn_wmma_f32_16x16x4_f32(
                false, a[i], false, bfrag, (short)0, acc, false, false);
        }

        // Scatter-add: element r of acc is (m = r + hi*8, n = ncol).
#pragma unroll
        for (int r = 0; r < 8; ++r) {
            if (drow[r] >= 0)
                atomicAdd(&msg_sum[(size_t)drow[r] * D_FEAT + ncol], acc[r]);
        }
    }
}

// ---------------------------------------------------------------------------
// Apply kernel: out = relu([h || msg_sum/deg] @ W_apply^T + b_apply)
// One wave per 16 nodes, same WMMA tiling.
// ---------------------------------------------------------------------------
__global__ void __launch_bounds__(256)
sage_apply_kernel(const float* __restrict__ nfeats,
                  const float* __restrict__ msg_sum,
                  const float* __restrict__ deg,
                  const float* __restrict__ Wa,   // 64 x 128 row-major
                  const float* __restrict__ ba,   // 64
                  float*       __restrict__ out,  // n_nodes x 64
                  int n_nodes)
{
    __shared__ float wlds[D_FEAT * LDS_WSTR];
    for (int i = threadIdx.x; i < D_FEAT * D_K; i += blockDim.x)
        wlds[(i >> 7) * LDS_WSTR + (i & 127)] = Wa[i];
    __syncthreads();

    const int wave = threadIdx.x >> 5;
    const int lane = threadIdx.x & 31;
    const int tile = blockIdx.x * WAVES_PB + wave;
    const int n0b  = tile * TILE_M;
    if (n0b >= n_nodes) return;

    const int row  = lane & 15;
    const int hi   = lane >> 4;
    const int koff = hi * 2;

    int node = n0b + row;
    bool row_valid = (node < n_nodes);
    if (!row_valid) node = n_nodes - 1;

    const float d = deg[node];
    const float scale = (d > 0.0f) ? (1.0f / d) : 0.0f;

    const float* hrow = nfeats  + (size_t)node * D_FEAT;
    const float* mrow = msg_sum + (size_t)node * D_FEAT;
    v2f a[32];
#pragma unroll
    for (int i = 0; i < 16; ++i)
        a[i] = *(const v2f*)(hrow + i * 4 + koff);
#pragma unroll
    for (int i = 0; i < 16; ++i) {
        v2f t = *(const v2f*)(mrow + i * 4 + koff);
        a[16 + i] = (v2f){t.x * scale, t.y * scale};
    }
    if (!row_valid) {
#pragma unroll
        for (int i = 0; i < 32; ++i) a[i] = (v2f){0.0f, 0.0f};
    }

#pragma unroll
    for (int n0 = 0; n0 < 4; ++n0) {
        const int ncol = n0 * 16 + row;
        const float bias = ba[ncol];
        v8f acc = {bias, bias, bias, bias, bias, bias, bias, bias};

        const float* wrow = wlds + ncol * LDS_WSTR + koff;
#pragma unroll
        for (int i = 0; i < 32; ++i) {
            v2f bfrag = *(const v2f*)(wrow + i * 4);
            acc = __builtin_amdgcn_wmma_f32_16x16x4_f32(
                false, a[i], false, bfrag, (short)0, acc, false, false);
        }

#pragma unroll
        for (int r = 0; r < 8; ++r) {
            int m = r + hi * 8;
            int n = n0b + m;
            if (n < n_nodes)
                out[(size_t)n * D_FEAT + ncol] = fmaxf(acc[r], 0.0f);
        }
    }
}

// ---------------------------------------------------------------------------
extern "C" void kernel_launch(void* const* d_in, const int* in_sizes, int n_in,
                              void* d_out, int out_size, void* d_ws, size_t ws_size,
                              hipStream_t stream) {
    const float* nfeats = (const float*)d_in[0];
    const float* efeats = (const float*)d_in[1];
    const int*   src    = (const int*)  d_in[2];
    const int*   dst    = (const int*)  d_in[3];
    const float* Wm     = (const float*)d_in[4];
    const float* bm     = (const float*)d_in[5];
    const float* Wa     = (const float*)d_in[6];
    const float* ba     = (const float*)d_in[7];
    float*       out    = (float*)d_out;

    const int n_nodes = in_sizes[0] / D_FEAT;
    const int n_edges = in_sizes[2];

    float* msg_sum = (float*)d_ws;                       // n_nodes*64 floats
    float* deg     = msg_sum + (size_t)n_nodes * D_FEAT; // n_nodes floats

    // 1) zero accumulators
    {
        int total = n_nodes * D_FEAT + n_nodes;
        sage_init_kernel<<<1024, 256, 0, stream>>>((float*)d_ws, total);
    }

    // 2) edge messages + degree + scatter
    {
        int edges_per_block = TILE_M * WAVES_PB;  // 128
        int blocks = (n_edges + edges_per_block - 1) / edges_per_block;
        sage_edge_kernel<<<blocks, 256, 0, stream>>>(
            nfeats, efeats, src, dst, Wm, bm, msg_sum, deg, n_edges);
    }

    // 3) apply + relu
    {
        int nodes_per_block = TILE_M * WAVES_PB;  // 128
        int blocks = (n_nodes + nodes_per_block - 1) / nodes_per_block;
        sage_apply_kernel<<<blocks, 256, 0, stream>>>(
            nfeats, msg_sum, deg, Wa, ba, out, n_nodes);
    }
}